// Ours_msg_mlp_74062416053450
// MI455X (gfx1250) — compile-verified
//
#include <hip/hip_runtime.h>
#include <cstddef>
#include <cstdint>

// ---------------------------------------------------------------------------
// MI455X / gfx1250 implementation.
// Heavy lifting: fp32 WMMA GEMMs (V_WMMA_F32_16X16X4_F32) for the four
// 8192x8192 @ 8192x128 products + all 128-wide weight products.
// B tile stored transposed in LDS so each WMMA operand pair is one
// contiguous ds_load_b64; each wave computes two 16-row bands to halve
// LDS traffic per WMMA. Scatter phases use native fp32 global atomics.
// ---------------------------------------------------------------------------

typedef __attribute__((ext_vector_type(2))) float v2f;
typedef __attribute__((ext_vector_type(8))) float v8f;

#define DF     128   // feature dim (DV == DE == 128)
#define KSTEP  32
#define PAD    2     // LDS row pad: stride 34 dwords -> conflict-free (64 banks)
#define BAND   2     // 16-row bands per wave
#define TILE_M (8 * 16 * BAND)   // 256 rows per block (8 waves)

__device__ __forceinline__ float atomAddF(float* p, float v) {
  return unsafeAtomicAdd(p, v);   // -> global_atomic_add_f32
}

// --------------------------- zero fill -------------------------------------
__global__ void k_zero(float* __restrict__ p, size_t n) {
  size_t i = (size_t)blockIdx.x * blockDim.x + threadIdx.x;
  size_t stride = (size_t)gridDim.x * blockDim.x;
  for (; i < n; i += stride) p[i] = 0.f;
}

// --------------------------- weighted scatter-add --------------------------
// out[dst[i]] += (w_src[src[i]] / s_dst[dst[i]]) * feat[src[i]]   (128-wide)
// One wave per nonzero; each lane handles one float4 (4 columns).
__global__ void k_scatter(const float* __restrict__ feat,
                          const float* __restrict__ w_src,
                          const float* __restrict__ s_dst,
                          const int* __restrict__ src,
                          const int* __restrict__ dst,
                          float* __restrict__ out, int nnz) {
  int i = blockIdx.x * 8 + (threadIdx.x >> 5);
  if (i >= nnz) return;
  int lane = threadIdx.x & 31;
  int s = src[i], d = dst[i];
  float w = w_src[s] / s_dst[d];
  float4 v = ((const float4*)(feat + (size_t)s * DF))[lane];
  float* o = out + (size_t)d * DF + lane * 4;
  atomAddF(o + 0, w * v.x);
  atomAddF(o + 1, w * v.y);
  atomAddF(o + 2, w * v.z);
  atomAddF(o + 3, w * v.w);
}

// --------------------------- fp32 WMMA GEMM --------------------------------
// C[M x 128] = A[M x K] @ B[K x 128]  (B row-major, ldb = 128)
// Block: 256 threads = 8 waves. Wave w computes rows [w*32, w*32+32) x all
// 128 cols: 2 bands x 8 accumulators of 16x16. grid.y = split-K factor; if
// > 1 the epilogue atomically accumulates into pre-zeroed C (bias must be
// null in that case).
__global__ __launch_bounds__(256)
void k_gemm(const float* __restrict__ A, const float* __restrict__ B,
            const float* __restrict__ bias, float* __restrict__ C,
            int M, int K, int lda) {
  __shared__ float As[TILE_M][KSTEP + PAD];   // row-major A tile
  __shared__ float Bst[DF][KSTEP + PAD];      // TRANSPOSED B tile: Bst[n][k]
  const int t    = threadIdx.x;
  const int wave = t >> 5;
  const int lane = t & 31;
  const int half = lane >> 4;
  const int j    = lane & 15;
  const int row0 = blockIdx.x * TILE_M;
  const int kChunk = K / gridDim.y;
  const int kBeg   = blockIdx.y * kChunk;

  v8f acc[BAND][8];
  const v8f vzero = {0.f, 0.f, 0.f, 0.f, 0.f, 0.f, 0.f, 0.f};
#pragma unroll
  for (int bq = 0; bq < BAND; ++bq)
#pragma unroll
    for (int g = 0; g < 8; ++g) acc[bq][g] = vzero;

  for (int k0 = kBeg; k0 < kBeg + kChunk; k0 += KSTEP) {
    // A tile: 256x32 floats, coalesced, streamed (non-temporal: read-once)
#pragma unroll
    for (int i = 0; i < 32; ++i) {
      int idx = t + i * 256;
      int r = idx >> 5, c = idx & 31;
      int gr = row0 + r;
      As[r][c] = (gr < M)
                     ? __builtin_nontemporal_load(&A[(size_t)gr * lda + (k0 + c)])
                     : 0.f;
    }
    // B tile: 32x128 coalesced global read, transposed LDS store
#pragma unroll
    for (int i = 0; i < 16; ++i) {
      int idx = t + i * 256;
      int r = idx >> 7, c = idx & 127;
      Bst[c][r] = B[(size_t)(k0 + r) * DF + c];
    }
    __syncthreads();

#pragma unroll
    for (int kk = 0; kk < KSTEP; kk += 4) {
      // B fragments: lane(j,half) holds B[K = kk+2h, kk+2h+1][N = 16g+j]
      // -> contiguous pair in Bst: single ds_load_b64 each
      v2f b[8];
#pragma unroll
      for (int g = 0; g < 8; ++g)
        b[g] = *(const v2f*)&Bst[g * 16 + j][kk + 2 * half];
#pragma unroll
      for (int bq = 0; bq < BAND; ++bq) {
        // A fragment: lane(j,half) holds A[M=j][K = kk+2h, kk+2h+1]
        v2f a = *(const v2f*)&As[wave * (16 * BAND) + bq * 16 + j][kk + 2 * half];
#pragma unroll
        for (int g = 0; g < 8; ++g) {
          acc[bq][g] = __builtin_amdgcn_wmma_f32_16x16x4_f32(
              false, a, false, b[g], (short)0, acc[bq][g], false, false);
        }
      }
    }
    __syncthreads();
  }

  // Epilogue. C/D layout: VGPR r of tile g -> C[m][n], m = r + 8*half, n = 16g+j
  const bool atomic = (gridDim.y > 1);
#pragma unroll
  for (int bq = 0; bq < BAND; ++bq) {
#pragma unroll
    for (int g = 0; g < 8; ++g) {
      const int n = g * 16 + j;
      const float bn = (bias != nullptr) ? bias[n] : 0.f;
#pragma unroll
      for (int r = 0; r < 8; ++r) {
        int m = row0 + wave * (16 * BAND) + bq * 16 + r + 8 * half;
        if (m < M) {
          float val = acc[bq][g][r];
          if (atomic) atomAddF(&C[(size_t)m * DF + n], val);
          else        C[(size_t)m * DF + n] = val + bn;
        }
      }
    }
  }
}

// --------------------------- row LayerNorm ---------------------------------
// out[row] = LN(relu?(X[row])) * gamma + beta (+ add[row])    (128 wide)
__global__ void k_ln(const float* __restrict__ X, const float* __restrict__ add,
                     const float* __restrict__ gamma, const float* __restrict__ beta,
                     float* __restrict__ out, int relu) {
  const int row = blockIdx.x;
  const int t = threadIdx.x;  // 128
  __shared__ float sh[DF];
  float x = X[(size_t)row * DF + t];
  if (relu) x = fmaxf(x, 0.f);
  sh[t] = x;
  __syncthreads();
  for (int s = 64; s > 0; s >>= 1) { if (t < s) sh[t] += sh[t + s]; __syncthreads(); }
  float mu = sh[0] * (1.f / DF);
  __syncthreads();
  float d = x - mu;
  sh[t] = d * d;
  __syncthreads();
  for (int s = 64; s > 0; s >>= 1) { if (t < s) sh[t] += sh[t + s]; __syncthreads(); }
  float var = sh[0] * (1.f / DF);
  float y = d * rsqrtf(var + 1e-5f) * gamma[t] + beta[t];
  if (add) y += add[(size_t)row * DF + t];
  out[(size_t)row * DF + t] = y;
}

// --------------------------- cosine-mean reductions ------------------------
// red[0] += cos(s[row], t[row]); red[1] += cos(s[row], s[perm1[row]]);
// red[2] += cos(t[row], t[perm2[row]])
__global__ void k_cos(const float* __restrict__ S, const float* __restrict__ T,
                      const int* __restrict__ perm1, const int* __restrict__ perm2,
                      float* __restrict__ red) {
  const int row = blockIdx.x;
  const int t = threadIdx.x;  // 128
  const float* sr = S + (size_t)row * DF;
  const float* tr = T + (size_t)row * DF;
  const float* sp = S + (size_t)perm1[row] * DF;
  const float* tp = T + (size_t)perm2[row] * DF;
  float a = sr[t], c = tr[t], d = sp[t], e = tp[t];
  __shared__ float sh[7][DF];
  sh[0][t] = a * c;  sh[1][t] = a * a;  sh[2][t] = c * c;
  sh[3][t] = a * d;  sh[4][t] = d * d;
  sh[5][t] = c * e;  sh[6][t] = e * e;
  __syncthreads();
  for (int s = 64; s > 0; s >>= 1) {
    if (t < s)
#pragma unroll
      for (int q = 0; q < 7; ++q) sh[q][t] += sh[q][t + s];
    __syncthreads();
  }
  if (t == 0) {
    const float eps = 1e-8f;
    float pos = sh[0][0] / fmaxf(sqrtf(sh[1][0]) * sqrtf(sh[2][0]), eps);
    float n1  = sh[3][0] / fmaxf(sqrtf(sh[1][0]) * sqrtf(sh[4][0]), eps);
    float n2  = sh[5][0] / fmaxf(sqrtf(sh[2][0]) * sqrtf(sh[6][0]), eps);
    atomAddF(&red[0], pos);
    atomAddF(&red[1], n1);
    atomAddF(&red[2], n2);
  }
}

// ssl_loss = log(sig(pos)) + log(1-sig(neg1)) + log(1-sig(neg2))
__global__ void k_loss(const float* __restrict__ red, float* __restrict__ out,
                       float invE) {
  if (threadIdx.x == 0 && blockIdx.x == 0) {
    float pos = red[0] * invE, n1 = red[1] * invE, n2 = red[2] * invE;
    // log(sigmoid(x)) = -log1p(exp(-x));  log(1-sigmoid(x)) = -log1p(exp(x))
    out[0] = -log1pf(expf(-pos)) - log1pf(expf(n1)) - log1pf(expf(n2));
  }
}

__global__ void k_mix(const float* __restrict__ a, const float* __restrict__ b,
                      float* __restrict__ o, float ratio, int n) {
  int i = blockIdx.x * 256 + threadIdx.x;
  if (i < n) o[i] = a[i] * ratio + b[i] * (1.f - ratio);
}

// ---------------------------------------------------------------------------
extern "C" void kernel_launch(void* const* d_in, const int* in_sizes, int n_in,
                              void* d_out, int out_size, void* d_ws, size_t ws_size,
                              hipStream_t stream) {
  const float* vfeat        = (const float*)d_in[0];
  const float* efeat        = (const float*)d_in[1];
  const float* v_reg_weight = (const float*)d_in[2];
  const float* v_reg_sum    = (const float*)d_in[3];
  const float* e_reg_weight = (const float*)d_in[4];
  const float* e_reg_sum    = (const float*)d_in[5];
  const float* A_struct     = (const float*)d_in[6];
  const float* A_temp       = (const float*)d_in[7];
  const int*   g1_src       = (const int*)d_in[8];
  const int*   g1_dst       = (const int*)d_in[9];
  const int*   g2_src       = (const int*)d_in[10];
  const int*   g2_dst       = (const int*)d_in[11];
  const int*   perm1        = (const int*)d_in[12];
  const int*   perm2        = (const int*)d_in[13];
  // d_in[14], d_in[15]: W_vtx / b_vtx -> dead code in reference (DCE'd)
  const float* W_ve  = (const float*)d_in[16];
  const float* b_ve  = (const float*)d_in[17];
  const float* ln_e_g = (const float*)d_in[18];
  const float* ln_e_b = (const float*)d_in[19];
  const float* W_ef  = (const float*)d_in[20];
  const float* b_ef  = (const float*)d_in[21];
  const float* sW1 = (const float*)d_in[22];
  const float* sb1 = (const float*)d_in[23];
  const float* sln_g = (const float*)d_in[24];
  const float* sln_b = (const float*)d_in[25];
  const float* sW2 = (const float*)d_in[26];
  const float* sb2 = (const float*)d_in[27];
  const float* tW1 = (const float*)d_in[28];
  const float* tb1 = (const float*)d_in[29];
  const float* tln_g = (const float*)d_in[30];
  const float* tln_b = (const float*)d_in[31];
  const float* tW2 = (const float*)d_in[32];
  const float* tb2 = (const float*)d_in[33];
  const float* W_ev  = (const float*)d_in[34];
  const float* b_ev  = (const float*)d_in[35];
  const float* ln_v_g = (const float*)d_in[36];
  const float* ln_v_b = (const float*)d_in[37];

  const int N   = in_sizes[0] / DF;   // 50000
  const int E   = in_sizes[1] / DF;   // 8192
  const int NNZ = in_sizes[8];        // 1000000

  // -------- workspace layout (floats) --------
  float* Wp = (float*)d_ws;
  size_t off = 0;
  float* norm_vfeat = Wp + off; off += (size_t)E * DF;   // zeroed prefix...
  float* norm_efeat = Wp + off; off += (size_t)N * DF;
  float* red        = Wp + off; off += 16;               // ...up to here
  float* bufP   = Wp + off; off += (size_t)E * DF;  // P / AX / AH (reused)
  float* bufQ   = Wp + off; off += (size_t)E * DF;  // Q / R (reused)
  float* efeat2 = Wp + off; off += (size_t)E * DF;
  float* bufH   = Wp + off; off += (size_t)E * DF;
  float* struct_e = Wp + off; off += (size_t)E * DF;
  float* temp_e   = Wp + off; off += (size_t)E * DF;
  float* bufRv  = Wp + off; off += (size_t)N * DF;

  float* out_feat_v = (float*)d_out;
  float* out_feat_e = (float*)d_out + (size_t)N * DF;
  float* out_loss   = (float*)d_out + (size_t)N * DF + (size_t)E * DF;

  auto gemm = [&](const float* A, const float* B, const float* bias, float* C,
                  int M, int K, int lda, int splitk) {
    dim3 grid((M + TILE_M - 1) / TILE_M, splitk);
    k_gemm<<<grid, 256, 0, stream>>>(A, B, bias, C, M, K, lda);
  };

  // zero accumulators (norm_vfeat, norm_efeat, red are contiguous at ws start)
  k_zero<<<2048, 256, 0, stream>>>(Wp, (size_t)E * DF + (size_t)N * DF + 16);

  // ---- g1: node -> hyperedge weighted aggregation ----
  k_scatter<<<(NNZ + 7) / 8, 256, 0, stream>>>(
      vfeat, v_reg_weight, e_reg_sum, g1_src, g1_dst, norm_vfeat, NNZ);

  // ---- efeat2 = LN(relu(norm_vfeat @ W_ve + b_ve)) + (efeat @ W_ef + b_ef) ----
  gemm(norm_vfeat, W_ve, b_ve, bufP, E, DF, DF, 1);
  gemm(efeat,      W_ef, b_ef, bufQ, E, DF, DF, 1);
  k_ln<<<E, DF, 0, stream>>>(bufP, bufQ, ln_e_g, ln_e_b, efeat2, 1);

  // ---- two dense GCNs (struct / temp) ----
  const float* As_[2] = {A_struct, A_temp};
  const float* W1_[2] = {sW1, tW1};
  const float* B1_[2] = {sb1, tb1};
  const float* LG_[2] = {sln_g, tln_g};
  const float* LB_[2] = {sln_b, tln_b};
  const float* W2_[2] = {sW2, tW2};
  const float* B2_[2] = {sb2, tb2};
  float* outs_[2] = {struct_e, temp_e};
  for (int g = 0; g < 2; ++g) {
    // AX = A @ efeat2   (split-K=8, atomic accumulate)
    k_zero<<<1024, 256, 0, stream>>>(bufP, (size_t)E * DF);
    gemm(As_[g], efeat2, nullptr, bufP, E, E, E, 8);
    // H = LN(AX @ W1 + b1)
    gemm(bufP, W1_[g], B1_[g], bufQ, E, DF, DF, 1);
    k_ln<<<E, DF, 0, stream>>>(bufQ, nullptr, LG_[g], LB_[g], bufH, 0);
    // out = (A @ H) @ W2 + b2
    k_zero<<<1024, 256, 0, stream>>>(bufP, (size_t)E * DF);
    gemm(As_[g], bufH, nullptr, bufP, E, E, E, 8);
    gemm(bufP, W2_[g], B2_[g], outs_[g], E, DF, DF, 1);
  }

  // ---- SSL loss ----
  k_cos<<<E, DF, 0, stream>>>(struct_e, temp_e, perm1, perm2, red);
  k_loss<<<1, 32, 0, stream>>>(red, out_loss, 1.f / (float)E);

  // ---- feat_e = 0.5*struct + 0.5*temp ----
  k_mix<<<(E * DF + 255) / 256, 256, 0, stream>>>(struct_e, temp_e, out_feat_e,
                                                  0.5f, E * DF);

  // ---- g2: hyperedge -> node weighted aggregation ----
  k_scatter<<<(NNZ + 7) / 8, 256, 0, stream>>>(
      out_feat_e, e_reg_weight, v_reg_sum, g2_src, g2_dst, norm_efeat, NNZ);

  // ---- feat_v = LN(relu(norm_efeat @ W_ev + b_ev)) ----
  gemm(norm_efeat, W_ev, b_ev, bufRv, N, DF, DF, 1);
  k_ln<<<N, DF, 0, stream>>>(bufRv, nullptr, ln_v_g, ln_v_b, out_feat_v, 1);
}